// NodeToWordsLayer_62251255988285
// MI455X (gfx1250) — compile-verified
//
#include <hip/hip_runtime.h>

// NodeToWordsLayer gather on MI455X (gfx1250, wave32).
//
// out[n, m, :] = (m < word_count(n)) ? x[clip(start_eff(n)+m, 0, S-1), :] : 0
//   word_count = (end==-1) ? 1 : end-start+1
//   start_eff  = start<0 ? start+S : start
//
// Zero-FLOP workload: bounded by the 410 MB streaming write (23.3 TB/s ->
// ~18us floor); the 10 MB source table is L2-resident. Design:
//   - one wave32 per NODE (8 rows = 8 KB): node math + record fetch paid once
//   - node record forced onto the scalar path (readfirstlane -> s_load_b64)
//   - fully unrolled m-loop: up to 16 in-flight global_load_b128 per lane
//   - non-temporal b128 stores so the write stream doesn't evict the table
//   - 32-bit element offsets (output 409.6 MB < 2 GB)

typedef __attribute__((ext_vector_type(4))) float v4f;

__global__ __launch_bounds__(256) void node_to_words_gather(
    const int* __restrict__ nodes,   // [N,2] int32 (start, end)
    const float* __restrict__ x,     // [S,256] float32
    float* __restrict__ out,         // [N,8,256] float32
    int N, int S)
{
    const int lane = threadIdx.x & 31;                 // lane within wave32
    const int wave = threadIdx.x >> 5;                 // 8 waves / 256-thread block
    int node = (int)blockIdx.x * 8 + wave;             // one node per wave
    if (node >= N) return;

    // Wave-uniform by construction; readfirstlane makes it provable so the
    // (start,end) pair becomes a single scalar s_load_b64 (KMcnt path).
    node = __builtin_amdgcn_readfirstlane(node);

    const int start = nodes[2 * node];
    const int end   = nodes[2 * node + 1];

    const int wc        = (end == -1) ? 1 : (end - start + 1);   // scalar
    const int start_eff = (start < 0) ? (start + S) : start;     // scalar

    // 32-bit element offsets: max out offset = N*2048 = 102.4M floats.
    v4f* __restrict__ dst =
        (v4f*)(out + (unsigned)node * 2048u) + lane;   // +lane: one v-address,
                                                       // m handled via IOFFSET

    // Warm the first gathered row while remaining address math resolves.
    {
        int idx0 = start_eff;
        idx0 = idx0 < 0 ? 0 : (idx0 > S - 1 ? S - 1 : idx0);
        __builtin_prefetch(x + (unsigned)idx0 * 256u + (unsigned)lane * 4u, 0, 3);
    }

    const v4f zero = {0.0f, 0.0f, 0.0f, 0.0f};

#pragma unroll
    for (int m = 0; m < 8; ++m) {
        v4f a, b;
        if (m < wc) {                                  // scalar branch
            int idx = start_eff + m;
            idx = idx < 0 ? 0 : (idx > S - 1 ? S - 1 : idx);
            const v4f* __restrict__ src =
                (const v4f*)(x + (unsigned)idx * 256u);
            a = src[lane];                             // global_load_b128
            b = src[lane + 32];                        // global_load_b128
        } else {
            a = zero;
            b = zero;
        }
        // Write-once output: non-temporal keeps L2 for the gather table.
        __builtin_nontemporal_store(a, dst + m * 64);
        __builtin_nontemporal_store(b, dst + m * 64 + 32);
    }
}

extern "C" void kernel_launch(void* const* d_in, const int* in_sizes, int n_in,
                              void* d_out, int out_size, void* d_ws, size_t ws_size,
                              hipStream_t stream) {
    const int*   nodes = (const int*)d_in[0];    // (1, N, 2) int32
    const float* x     = (const float*)d_in[1];  // (1, S, 256) float32
    float*       out   = (float*)d_out;          // (1, N, 8, 256) float32

    const int N = in_sizes[0] / 2;               // 50000
    const int S = in_sizes[1] / 256;             // 10000

    const int blocks = (N + 7) / 8;              // one wave32 per node
    node_to_words_gather<<<blocks, 256, 0, stream>>>(nodes, x, out, N, S);
}